// HSAAttention_30262339567741
// MI455X (gfx1250) — compile-verified
//
#include <hip/hip_runtime.h>
#include <math.h>

typedef float v2f __attribute__((ext_vector_type(2)));
typedef float v8f __attribute__((ext_vector_type(8)));

#define WMMA_F32(a, b, c) \
    __builtin_amdgcn_wmma_f32_16x16x4_f32(false, (a), false, (b), (short)0, (c), false, false)

// ---- CDNA5 async global->LDS staging (fallback to float4 if not declared) --
#if defined(__AMDGCN__)
#if __has_builtin(__builtin_amdgcn_global_load_async_to_lds_b128) && \
    __has_builtin(__builtin_amdgcn_global_load_async_to_lds_b32) &&  \
    __has_builtin(__builtin_amdgcn_s_wait_asynccnt)
#define HSA_ASYNC_LDS 1
#endif
#endif

#ifdef HSA_ASYNC_LDS
// builtin wants typed pointers: v4i addrspace(1)* (global) / addrspace(3)* (LDS)
typedef int hsa_v4i __attribute__((vector_size(16)));
typedef __attribute__((address_space(1))) hsa_v4i* hsa_g128_t;
typedef __attribute__((address_space(3))) hsa_v4i* hsa_l128_t;
typedef __attribute__((address_space(1))) int* hsa_g32_t;
typedef __attribute__((address_space(3))) int* hsa_l32_t;
#define ASYNC_COPY_B128(g, l)                                                \
    __builtin_amdgcn_global_load_async_to_lds_b128((hsa_g128_t)(g),          \
                                                   (hsa_l128_t)(l), 0, 0)
#define ASYNC_COPY_B32(g, l)                                                 \
    __builtin_amdgcn_global_load_async_to_lds_b32((hsa_g32_t)(g),            \
                                                  (hsa_l32_t)(l), 0, 0)
#define ASYNC_WAIT() __builtin_amdgcn_s_wait_asynccnt(0)
#else
#define ASYNC_COPY_B128(g, l) (*(float4*)(l) = *(const float4*)(g))
#define ASYNC_COPY_B32(g, l) (*(float*)(l) = *(const float*)(g))
#define ASYNC_WAIT() ((void)0)
#endif

static __device__ inline v8f v8f_zero() {
    v8f z;
#pragma unroll
    for (int i = 0; i < 8; ++i) z[i] = 0.f;
    return z;
}

// ---------------------------------------------------------------------------
// Generic 4096x1024x1024 GEMM: out = A @ W + bias, with epilogue modes:
//   mode 0: scale by Dh^-0.5 and scatter to [B*H][S][Dh] head layout (Q)
//   mode 1: scatter to head layout (V)
//   mode 2: plain [4096][1024] layout (output projection)
// 256 threads (8 waves), tile 64x64, k-chunk 16, double-buffered async
// global->LDS staging overlapped with WMMA compute (one barrier per chunk).
// ---------------------------------------------------------------------------
__global__ __launch_bounds__(256) void gemm_nt(const float* __restrict__ A,
                                               const float* __restrict__ W,
                                               const float* __restrict__ bias,
                                               float* __restrict__ out, int mode) {
    __shared__ float As[2][64 * 16];
    __shared__ float Bs[2][16 * 64];
    const int tid = threadIdx.x;
    const int lane = tid & 31, wv = tid >> 5;
    const int nl = lane & 15, half = lane >> 4, koff = half * 2;
    const int m0 = blockIdx.y * 64, n0 = blockIdx.x * 64;
    const int mt = wv & 3, ng = wv >> 2;  // wave -> 16-row tile, 32-col group
    const int ra = tid >> 2, ca = (tid & 3) * 4;   // A-stage: 64x16, float4/lane
    const int rb = tid >> 4, cb = (tid & 15) * 4;  // B-stage: 16x64, float4/lane

    v8f acc0 = v8f_zero(), acc1 = v8f_zero();

    // prologue: issue chunk 0
    ASYNC_COPY_B128(A + (m0 + ra) * 1024 + ca, &As[0][ra * 16 + ca]);
    ASYNC_COPY_B128(W + rb * 1024 + n0 + cb, &Bs[0][rb * 64 + cb]);

    int buf = 0;
    for (int kc = 0; kc < 1024; kc += 16) {
        ASYNC_WAIT();        // our async writes into LDS have landed
        __syncthreads();     // everyone's have landed
        if (kc + 16 < 1024) {  // overlap next chunk's copy with compute
            ASYNC_COPY_B128(A + (m0 + ra) * 1024 + (kc + 16) + ca,
                            &As[buf ^ 1][ra * 16 + ca]);
            ASYNC_COPY_B128(W + (kc + 16 + rb) * 1024 + n0 + cb,
                            &Bs[buf ^ 1][rb * 64 + cb]);
        }
        const float* as = As[buf];
        const float* bs = Bs[buf];
#pragma unroll
        for (int kk = 0; kk < 16; kk += 4) {
            v2f a;
            a.x = as[(mt * 16 + nl) * 16 + kk + koff];
            a.y = as[(mt * 16 + nl) * 16 + kk + koff + 1];
            v2f b0;
            b0.x = bs[(kk + koff) * 64 + ng * 32 + nl];
            b0.y = bs[(kk + koff + 1) * 64 + ng * 32 + nl];
            acc0 = WMMA_F32(a, b0, acc0);
            v2f b1;
            b1.x = bs[(kk + koff) * 64 + ng * 32 + 16 + nl];
            b1.y = bs[(kk + koff + 1) * 64 + ng * 32 + 16 + nl];
            acc1 = WMMA_F32(a, b1, acc1);
        }
        buf ^= 1;
    }

#pragma unroll
    for (int r = 0; r < 8; ++r) {
        const int row = m0 + mt * 16 + r + 8 * half;
        const int c0 = n0 + ng * 32 + nl;
        const int c1 = c0 + 16;
        float v0 = acc0[r] + bias[c0];
        float v1 = acc1[r] + bias[c1];
        if (mode == 2) {
            out[row * 1024 + c0] = v0;
            out[row * 1024 + c1] = v1;
        } else {
            if (mode == 0) { v0 *= 0.125f; v1 *= 0.125f; }  // Dh^-0.5
            const int b = row >> 10, s = row & 1023;
            out[((b * 16 + (c0 >> 6)) * 1024 + s) * 64 + (c0 & 63)] = v0;
            out[((b * 16 + (c1 >> 6)) * 1024 + s) * 64 + (c1 & 63)] = v1;
        }
    }
}

// ---------------------------------------------------------------------------
// Psi[bh][p][s] = sqrt(w_p) * exp(-0.5 * (q-mu_p)^T P_p (q-mu_p))
// Per block: one splat p, 64 tokens. Y = D @ P_p via WMMA f32, then
// m = rowdot(D, Y) via elementwise product in C-layout + 16-lane butterfly.
// ---------------------------------------------------------------------------
__global__ __launch_bounds__(128) void psi_kernel(const float* __restrict__ Q,
                                                  const float* __restrict__ mu,
                                                  const float* __restrict__ prec,
                                                  const float* __restrict__ sw,
                                                  float* __restrict__ Psi) {
    __shared__ float Pt[64 * 64];
    __shared__ float Qt[64 * 64];
    __shared__ float muS[64];
    const int tid = threadIdx.x;
    const int s0 = blockIdx.x * 64, bh = blockIdx.y, p = blockIdx.z;

    for (int i = tid; i < 1024; i += 128) {
        ASYNC_COPY_B128(prec + p * 4096 + i * 4, Pt + i * 4);
        ASYNC_COPY_B128(Q + (bh * 1024 + s0) * 64 + i * 4, Qt + i * 4);
    }
    if (tid < 64) muS[tid] = mu[p * 64 + tid];
    ASYNC_WAIT();
    __syncthreads();

    const float sqw = sqrtf(sw[p]);
    const int lane = tid & 31, wv = tid >> 5;
    const int nl = lane & 15, half = lane >> 4, koff = half * 2;
    const int mrow = wv * 16 + nl;

    v8f acc[4];
#pragma unroll
    for (int c = 0; c < 4; ++c) acc[c] = v8f_zero();

#pragma unroll
    for (int kk = 0; kk < 64; kk += 4) {
        v2f a;
        a.x = Qt[mrow * 64 + kk + koff] - muS[kk + koff];
        a.y = Qt[mrow * 64 + kk + koff + 1] - muS[kk + koff + 1];
#pragma unroll
        for (int c = 0; c < 4; ++c) {
            v2f b;
            b.x = Pt[(kk + koff) * 64 + c * 16 + nl];
            b.y = Pt[(kk + koff + 1) * 64 + c * 16 + nl];
            acc[c] = WMMA_F32(a, b, acc[c]);
        }
    }

    float s[8];
#pragma unroll
    for (int r = 0; r < 8; ++r) s[r] = 0.f;
#pragma unroll
    for (int c = 0; c < 4; ++c) {
        const int col = c * 16 + nl;
        const float mv = muS[col];
#pragma unroll
        for (int r = 0; r < 8; ++r) {
            const float d = Qt[(wv * 16 + r + 8 * half) * 64 + col] - mv;
            s[r] += d * acc[c][r];
        }
    }
#pragma unroll
    for (int r = 0; r < 8; ++r) {  // reduce across the 16-lane half-group
        s[r] += __shfl_xor(s[r], 1, 32);
        s[r] += __shfl_xor(s[r], 2, 32);
        s[r] += __shfl_xor(s[r], 4, 32);
        s[r] += __shfl_xor(s[r], 8, 32);
    }
#pragma unroll
    for (int r = 0; r < 8; ++r) {
        if (nl == r) {
            const int srow = s0 + wv * 16 + r + 8 * half;
            Psi[(bh * 112 + p) * 1024 + srow] = sqw * __expf(-0.5f * s[r]);
        }
    }
}

// ---------------------------------------------------------------------------
// Fused attention: per block handles one (b,h) and 32 query rows.
//  1) attn[32][1024] = PsiRows^T @ Psi  (K=112) -> LDS (128 KB slab)
//  2) exact top-64 threshold per row via bitwise binary search, mask, norm
//  3) ctx[32][64] = attn @ V_bh          (K=1024)
// attn never touches HBM. 256 threads = 8 waves.
// ---------------------------------------------------------------------------
__global__ __launch_bounds__(256) void attn_kernel(const float* __restrict__ Psi,
                                                   const float* __restrict__ V,
                                                   const float* __restrict__ amask,
                                                   float* __restrict__ Ctx) {
    __shared__ float attnS[32 * 1024];  // 128 KB
    __shared__ float As[112 * 32];      // Psi rows (transposed A operand)
    __shared__ float maskS[1024];
    const int tid = threadIdx.x;
    const int lane = tid & 31, wv = tid >> 5;
    const int nl = lane & 15, half = lane >> 4, koff = half * 2;
    const int i0 = blockIdx.x * 32, bh = blockIdx.y;
    const int b = bh >> 4, h = bh & 15;
    const float* PsiB = Psi + bh * 112 * 1024;
    const float* Vb = V + bh * 1024 * 64;

    for (int idx = tid; idx < 112 * 32; idx += 256)
        ASYNC_COPY_B32(PsiB + (idx >> 5) * 1024 + i0 + (idx & 31), As + idx);
    ASYNC_COPY_B128(amask + b * 1024 + tid * 4, maskS + tid * 4);
    // warm the stage-3 V panel (256 KB) while stages 1-2 run
    for (int j = tid; j < 2048; j += 256) __builtin_prefetch(Vb + j * 32, 0, 0);
    ASYNC_WAIT();
    __syncthreads();

    // --- stage 1: attn slab (each wave: 8 strips of 16 cols x 32 rows) ---
    for (int strip = 0; strip < 8; ++strip) {
        const int n0 = wv * 16 + strip * 128;
        v8f a0 = v8f_zero(), a1 = v8f_zero();
#pragma unroll
        for (int kk = 0; kk < 112; kk += 4) {
            v2f bm;  // B from L2-resident Psi
            bm.x = PsiB[(kk + koff) * 1024 + n0 + nl];
            bm.y = PsiB[(kk + koff + 1) * 1024 + n0 + nl];
            v2f av0, av1;
            av0.x = As[(kk + koff) * 32 + nl];
            av0.y = As[(kk + koff + 1) * 32 + nl];
            av1.x = As[(kk + koff) * 32 + 16 + nl];
            av1.y = As[(kk + koff + 1) * 32 + 16 + nl];
            a0 = WMMA_F32(av0, bm, a0);
            a1 = WMMA_F32(av1, bm, a1);
        }
#pragma unroll
        for (int r = 0; r < 8; ++r) {
            attnS[(r + 8 * half) * 1024 + n0 + nl] = a0[r];
            attnS[(16 + r + 8 * half) * 1024 + n0 + nl] = a1[r];
        }
    }
    __syncthreads();

    // --- stage 2: exact top-64 threshold (positive floats: bit search) ---
    for (int rr = wv; rr < 32; rr += 8) {
        float regs[32];
#pragma unroll
        for (int i = 0; i < 32; ++i) regs[i] = attnS[rr * 1024 + lane + 32 * i];
        unsigned res = 0u;
        for (int bit = 30; bit >= 0; --bit) {
            const unsigned cand = res | (1u << bit);
            const float cf = __uint_as_float(cand);
            int cnt = 0;
#pragma unroll
            for (int i = 0; i < 32; ++i) cnt += (regs[i] >= cf) ? 1 : 0;
            cnt += __shfl_xor(cnt, 1, 32);
            cnt += __shfl_xor(cnt, 2, 32);
            cnt += __shfl_xor(cnt, 4, 32);
            cnt += __shfl_xor(cnt, 8, 32);
            cnt += __shfl_xor(cnt, 16, 32);
            if (cnt >= 64) res = cand;  // largest thr with count >= TOPK
        }
        const float thr = __uint_as_float(res);
        float sum = 0.f;
#pragma unroll
        for (int i = 0; i < 32; ++i) {
            float v = regs[i];
            v = (v >= thr) ? v : 0.f;
            v *= maskS[lane + 32 * i];
            regs[i] = v;
            sum += v;
        }
        sum += __shfl_xor(sum, 1, 32);
        sum += __shfl_xor(sum, 2, 32);
        sum += __shfl_xor(sum, 4, 32);
        sum += __shfl_xor(sum, 8, 32);
        sum += __shfl_xor(sum, 16, 32);
        const float inv = 1.f / fmaxf(sum, 1e-12f);
#pragma unroll
        for (int i = 0; i < 32; ++i) attnS[rr * 1024 + lane + 32 * i] = regs[i] * inv;
    }
    __syncthreads();

    // --- stage 3: ctx = attn @ V_bh (each wave owns one 16x16 tile) ---
    {
        const int mt = wv & 1, ct = wv >> 1;
        v8f acc = v8f_zero();
        for (int kk = 0; kk < 1024; kk += 4) {
            v2f a;
            a.x = attnS[(mt * 16 + nl) * 1024 + kk + koff];
            a.y = attnS[(mt * 16 + nl) * 1024 + kk + koff + 1];
            v2f bb;
            bb.x = Vb[(kk + koff) * 64 + ct * 16 + nl];
            bb.y = Vb[(kk + koff + 1) * 64 + ct * 16 + nl];
            acc = WMMA_F32(a, bb, acc);
        }
#pragma unroll
        for (int r = 0; r < 8; ++r) {
            const int srow = i0 + mt * 16 + r + 8 * half;
            Ctx[(b * 1024 + srow) * 1024 + h * 64 + ct * 16 + nl] = acc[r];
        }
    }
}

extern "C" void kernel_launch(void* const* d_in, const int* in_sizes, int n_in,
                              void* d_out, int out_size, void* d_ws, size_t ws_size,
                              hipStream_t stream) {
    (void)in_sizes; (void)n_in; (void)out_size; (void)ws_size;
    const float* hidden = (const float*)d_in[0];
    const float* amask  = (const float*)d_in[1];
    const float* Wq = (const float*)d_in[2];
    const float* bq = (const float*)d_in[3];
    // d_in[4]/d_in[5] (Wk/bk) are dead in the reference — skipped.
    const float* Wv = (const float*)d_in[6];
    const float* bv = (const float*)d_in[7];
    const float* Wo = (const float*)d_in[8];
    const float* bo = (const float*)d_in[9];
    const float* mu = (const float*)d_in[10];
    const float* prec = (const float*)d_in[11];
    const float* sw = (const float*)d_in[12];
    float* out = (float*)d_out;

    float* Q   = (float*)d_ws;                 // [64][1024][64]  16 MB
    float* Vp  = Q + 4 * 1024 * 1024;          // [64][1024][64]  16 MB
    float* Psi = Vp + 4 * 1024 * 1024;         // [64][112][1024] 29.4 MB
    float* Ctx = Psi + 64 * 112 * 1024;        // [4][1024][1024] 16 MB

    const dim3 gg(16, 64), gb(256);
    gemm_nt<<<gg, gb, 0, stream>>>(hidden, Wq, bq, Q, 0);
    gemm_nt<<<gg, gb, 0, stream>>>(hidden, Wv, bv, Vp, 1);
    psi_kernel<<<dim3(16, 64, 112), 128, 0, stream>>>(Q, mu, prec, sw, Psi);
    attn_kernel<<<dim3(32, 64), 256, 0, stream>>>(Psi, Vp, amask, Ctx);
    gemm_nt<<<gg, gb, 0, stream>>>(Ctx, Wo, bo, out, 2);
}